// AOClassifierJax_37383395344743
// MI455X (gfx1250) — compile-verified
//
#include <hip/hip_runtime.h>
#include <stdint.h>

#ifndef __has_builtin
#define __has_builtin(x) 0
#endif

#if __has_builtin(__builtin_amdgcn_global_load_async_to_lds_b32)
#define ASYNC_LDS 1
#else
#define ASYNC_LDS 0
#endif

#if __has_builtin(__builtin_amdgcn_wmma_f32_16x16x4_f32)
#define HAVE_WMMA_F32X4 1
#else
#define HAVE_WMMA_F32X4 0
#endif

typedef float v8f __attribute__((ext_vector_type(8)));
typedef float v2f __attribute__((ext_vector_type(2)));

#define BATCH 4096
#define NCLS  512
#define DIM   256

#define BM 128        // batch rows per block
#define BN 64         // classes per block
#define KD 32         // K-chunk (== wave32 lanes; one lane per d)
#define TM 4          // micro-tile rows per thread
#define TN 8          // micro-tile cols per thread
#define BMP 132       // padded xs row (dwords): 16B-aligned b128, conflict-free
#define BNP 68        // padded ws row (dwords)
#define NTH 256       // 8 waves
#define NWAVE 8

#define SMEM_BYTES (2 * (KD * BMP + KD * BNP) * 4)   // 51200 B

typedef int __attribute__((address_space(3)))* lds_ip;
typedef int __attribute__((address_space(1)))* gbl_ip;

__device__ __forceinline__ lds_ip to_lds(const void* p) {
  // generic LDS pointer: low 32 bits are the LDS byte address (ISA 10.2 aperture map)
  return (lds_ip)(uint32_t)(uintptr_t)p;
}
__device__ __forceinline__ gbl_ip to_gbl(const void* p) {
  return (gbl_ip)(uintptr_t)p;
}

// Stage one K-chunk of x and W into LDS, TRANSPOSED to d-major:
// xs[d][b], ws[d][c].  Async path: lane = d, so each wave instruction reads a
// 128B-contiguous global row segment and scatters it down one LDS column (one dword in each of the 32 d rows).
__device__ __forceinline__ void stage_chunk(
    const float* __restrict__ x, const float* __restrict__ W,
    float (* __restrict__ xs)[BMP], float (* __restrict__ ws)[BNP],
    int row0, int col0, int d0, int tid)
{
#if ASYNC_LDS
  const int lane = tid & 31;
  const int wid  = tid >> 5;
  #pragma unroll
  for (int i = 0; i < BM / NWAVE; ++i) {
    const int bb = wid * (BM / NWAVE) + i;
    const float* src = x + (size_t)(row0 + bb) * DIM + d0 + lane;
    __builtin_amdgcn_global_load_async_to_lds_b32(to_gbl(src), to_lds(&xs[lane][bb]), 0, 0);
  }
  #pragma unroll
  for (int i = 0; i < BN / NWAVE; ++i) {
    const int cc = wid * (BN / NWAVE) + i;
    const float* src = W + (size_t)(col0 + cc) * DIM + d0 + lane;
    __builtin_amdgcn_global_load_async_to_lds_b32(to_gbl(src), to_lds(&ws[lane][cc]), 0, 0);
  }
#else
  for (int idx = tid; idx < BM * KD; idx += NTH) {
    const int bb = idx >> 5, d = idx & 31;
    xs[d][bb] = x[(size_t)(row0 + bb) * DIM + d0 + d];
  }
  for (int idx = tid; idx < BN * KD; idx += NTH) {
    const int cc = idx >> 5, d = idx & 31;
    ws[d][cc] = W[(size_t)(col0 + cc) * DIM + d0 + d];
  }
#endif
}

__device__ __forceinline__ void wait_stage() {
#if ASYNC_LDS
#if __has_builtin(__builtin_amdgcn_s_wait_asynccnt)
  __builtin_amdgcn_s_wait_asynccnt(0);
#else
  asm volatile("s_wait_asynccnt 0" ::: "memory");
#endif
#endif
}

__global__ __launch_bounds__(NTH) void l1_cls_kernel(
    const float* __restrict__ x, const float* __restrict__ W,
    const float* __restrict__ bias, float* __restrict__ out)
{
  __shared__ __align__(16) char smem[SMEM_BYTES];
  float (*xs0)[BMP] = (float (*)[BMP])smem;               // [KD][BMP] double-buffered
  float (*xs1)[BMP] = xs0 + KD;
  float (*ws0)[BNP] = (float (*)[BNP])(smem + 2 * KD * BMP * 4);
  float (*ws1)[BNP] = ws0 + KD;

  const int tid  = threadIdx.x;
  const int tx   = tid & 7;       // 8 col-threads  -> 8*TN = 64 cols
  const int ty   = tid >> 3;      // 32 row-threads -> 32*TM = 128 rows
  const int col0 = blockIdx.x * BN;
  const int row0 = blockIdx.y * BM;

  float acc[TM][TN];
  #pragma unroll
  for (int i = 0; i < TM; ++i)
    #pragma unroll
    for (int j = 0; j < TN; ++j) acc[i][j] = 0.0f;

  stage_chunk(x, W, xs0, ws0, row0, col0, 0, tid);

  const int nk = DIM / KD;   // 8 chunks
  for (int k = 0; k < nk; ++k) {
    wait_stage();            // my async writes for buf[k&1] have landed in LDS
    __syncthreads();         // everyone's have; everyone done reading other buf
    if (k + 1 < nk)
      stage_chunk(x, W, ((k + 1) & 1) ? xs1 : xs0, ((k + 1) & 1) ? ws1 : ws0,
                  row0, col0, (k + 1) * KD, tid);   // prefetch overlaps compute
    const float (*xsb)[BMP] = (k & 1) ? xs1 : xs0;
    const float (*wsb)[BNP] = (k & 1) ? ws1 : ws0;
    #pragma unroll 4
    for (int d = 0; d < KD; ++d) {
      const float4 xa = *(const float4*)&xsb[d][TM * ty];      // broadcast across tx
      const float4 w0 = *(const float4*)&wsb[d][TN * tx];      // conflict-free b128
      const float4 w1 = *(const float4*)&wsb[d][TN * tx + 4];
      const float xv[TM] = {xa.x, xa.y, xa.z, xa.w};
      const float wv[TN] = {w0.x, w0.y, w0.z, w0.w, w1.x, w1.y, w1.z, w1.w};
      #pragma unroll
      for (int i = 0; i < TM; ++i)
        #pragma unroll
        for (int j = 0; j < TN; ++j)
          acc[i][j] += __builtin_fabsf(wv[j] - xv[i]);   // v_sub + v_add |src|
    }
  }

  // ---- Epilogue: re-layout tile in LDS, add bias on the matrix unit ----
  // out = -dist + bias ; bias broadcast done as D = A x B + C with
  // A[m][k] = (k==0), B[0][n] = bias[n]  (exact: a single 1.0*bias product).
  __syncthreads();                               // all done reading xs/ws
  float (*tile)[BN + 4] = (float (*)[BN + 4])smem;   // 128 x 68 overlay
  #pragma unroll
  for (int i = 0; i < TM; ++i) {
    float4 o0 = make_float4(-acc[i][0], -acc[i][1], -acc[i][2], -acc[i][3]);
    float4 o1 = make_float4(-acc[i][4], -acc[i][5], -acc[i][6], -acc[i][7]);
    *(float4*)&tile[TM * ty + i][TN * tx]     = o0;
    *(float4*)&tile[TM * ty + i][TN * tx + 4] = o1;
  }
  __syncthreads();

  const int lane = tid & 31;
  const int wid  = tid >> 5;      // 8 waves -> 8 row bands of 16
  const int h    = lane >> 4;     // C-layout half: rows v (h=0) / v+8 (h=1)
  const int n    = lane & 15;     // column within 16x16 tile
  #pragma unroll
  for (int sc = 0; sc < BN / 16; ++sc) {
    v8f c;
    #pragma unroll
    for (int v = 0; v < 8; ++v)
      c[v] = tile[16 * wid + v + 8 * h][16 * sc + n];   // conflict-free gather
    const float bval = bias[col0 + 16 * sc + n];
#if HAVE_WMMA_F32X4
    v2f a  = { (h == 0) ? 1.0f : 0.0f, 0.0f };          // A: K=0 column of ones
    v2f bb = { (h == 0) ? bval : 0.0f, 0.0f };          // B: row K=0 = bias
    c = __builtin_amdgcn_wmma_f32_16x16x4_f32(
        /*neg_a=*/false, a, /*neg_b=*/false, bb,
        /*c_mod=*/(short)0, c, /*reuse_a=*/false, /*reuse_b=*/false);
#else
    #pragma unroll
    for (int v = 0; v < 8; ++v) c[v] += bval;
#endif
    #pragma unroll
    for (int v = 0; v < 8; ++v) {
      const int r = row0 + 16 * wid + v + 8 * h;
      out[(size_t)r * NCLS + (col0 + 16 * sc + n)] = c[v];
    }
  }
}

extern "C" void kernel_launch(void* const* d_in, const int* in_sizes, int n_in,
                              void* d_out, int out_size, void* d_ws, size_t ws_size,
                              hipStream_t stream) {
  (void)in_sizes; (void)n_in; (void)d_ws; (void)ws_size; (void)out_size;
  const float* x   = (const float*)d_in[0];
  const float* W   = (const float*)d_in[1];
  const float* b   = (const float*)d_in[2];
  float*       out = (float*)d_out;
  dim3 grid(NCLS / BN, BATCH / BM);   // (8, 32) = 256 workgroups
  l1_cls_kernel<<<grid, NTH, 0, stream>>>(x, W, b, out);
}